// my_GCN_14139032338889
// MI455X (gfx1250) — compile-verified
//
#include <hip/hip_runtime.h>

// GCN (3-layer, fp32) for MI455X / gfx1250.
// Dense per-node GEMMs ride V_WMMA_F32_16X16X4_F32 (exact fp32 precision,
// matches reference). Graph aggregation = L2-resident scatter-add atomics.
//
// Round-1 -> Round-2 change: weights staged in LDS k-pair interleaved so each
// WMMA B-fragment is one aligned 8-byte ds load (no v_mov repacking), matching
// the A-fragment path.

typedef __attribute__((ext_vector_type(2))) float v2f;
typedef __attribute__((ext_vector_type(8))) float v8f;

#define DIN 32
#define DH  48
#define DOUT 16

static __device__ __forceinline__ void atomic_add_f32(float* p, float v) {
    // relaxed agent-scope fp32 add -> global_atomic_add_f32 (no return)
    __hip_atomic_fetch_add(p, v, __ATOMIC_RELAXED, __HIP_MEMORY_SCOPE_AGENT);
}

// ---------------- degree / normalization ----------------

__global__ void k_deg_init(float* __restrict__ deg, int n) {
    int i = blockIdx.x * blockDim.x + threadIdx.x;
    if (i < n) deg[i] = 1.0f;   // self-loop contributes 1 to every node
}

__global__ void k_deg_count(const int* __restrict__ dst, float* __restrict__ deg, int e) {
    int i = blockIdx.x * blockDim.x + threadIdx.x;
    if (i < e) atomic_add_f32(&deg[dst[i]], 1.0f);
}

__global__ void k_rsqrt_inplace(float* __restrict__ deg, int n) {
    int i = blockIdx.x * blockDim.x + threadIdx.x;
    if (i < n) {
        float d = deg[i];
        deg[i] = (d > 0.0f) ? rsqrtf(d) : 0.0f;   // deg >= 1 here, guard for parity
    }
}

// ---------------- dense GEMM: hout[n,M] = act(hin[n,K]) @ W[K,M] ----------------
// Block = 128 threads = 4 waves; each wave owns a 16-row stripe (64 rows/block).
// WMMA: D(16x16,f32) += A(16x4,f32) x B(4x16,f32), K stepped by 4.
//   A layout (ISA 7.12.2): lanes 0-15 hold {K=k,k+1}, lanes 16-31 hold {K=k+2,k+3}.
//   B mirrored; C/D: VGPR v holds row (v + 8*(lane>=16)), col = lane&15.
// sWp holds W with k-pairs interleaved: sWp[k/2][n][0..1] = {W[k][n], W[k+1][n]}
// so a B fragment is a single aligned 8B LDS load.

template <int K, int M, bool RELU_IN>
__global__ __launch_bounds__(128)
void k_gemm_wmma(const float* __restrict__ hin, const float* __restrict__ W,
                 float* __restrict__ hout, int n) {
    __shared__ float sWp[K * M];       // (K/2) pair-rows x M cols x 2
    __shared__ float sH[64 * K];

    const int tid = threadIdx.x;
    const int rowBlock = blockIdx.x * 64;

    for (int i = tid; i < K * M; i += 128) {
        int k = i / M, c = i % M;
        sWp[(k >> 1) * (2 * M) + c * 2 + (k & 1)] = W[i];
    }
    for (int i = tid; i < 64 * K; i += 128) {
        int r = i / K, c = i % K;
        int g = rowBlock + r;
        float v = (g < n) ? hin[(size_t)g * K + c] : 0.0f;
        if (RELU_IN) v = fmaxf(v, 0.0f);           // ReLU of previous layer fused here
        sH[i] = v;
    }
    __syncthreads();

    const int wave = tid >> 5;
    const int lane = tid & 31;
    const int half = lane >> 4;        // 0 = lanes 0-15, 1 = lanes 16-31
    const int l15  = lane & 15;
    const int rbase = wave * 16;

    v8f acc[M / 16] = {};              // zero accumulators

#pragma unroll
    for (int k = 0; k < K; k += 4) {
        const int kk = k + 2 * half;               // per-lane-half K pair (even)
        v2f a = *(const v2f*)(sH + (rbase + l15) * K + kk);
        const int pairRow = (k >> 1) + half;       // == kk >> 1
#pragma unroll
        for (int t = 0; t < M / 16; ++t) {
            v2f b = *(const v2f*)(sWp + pairRow * (2 * M) + 2 * (t * 16 + l15));
            acc[t] = __builtin_amdgcn_wmma_f32_16x16x4_f32(
                false, a, false, b, (short)0, acc[t], false, false);
        }
    }

#pragma unroll
    for (int t = 0; t < M / 16; ++t) {
#pragma unroll
        for (int v = 0; v < 8; ++v) {
            int row = rowBlock + rbase + v + 8 * half;
            if (row < n) hout[(size_t)row * M + t * 16 + l15] = acc[t][v];
        }
    }
}

// ---------------- self-loop + bias init:  out = hW * dinv^2 + b ----------------

template <int M>
__global__ void k_self_bias(const float* __restrict__ hW, const float* __restrict__ dinv,
                            const float* __restrict__ bias, float* __restrict__ out, int n) {
    int gid = blockIdx.x * blockDim.x + threadIdx.x;
    if (gid < n * M) {
        int i = gid / M, c = gid % M;
        float di = dinv[i];
        out[gid] = hW[gid] * di * di + bias[c];
    }
}

// ---------------- edge scatter:  out[dst] += hW[src] * dinv[src]*dinv[dst] ----------------
// 16 lanes per edge; indices + norm loaded once per 16-lane group, channels strided.

template <int M>
__global__ void k_scatter(const int* __restrict__ src, const int* __restrict__ dst,
                          const float* __restrict__ hW, const float* __restrict__ dinv,
                          float* __restrict__ out, int e) {
    int gid = blockIdx.x * blockDim.x + threadIdx.x;
    int edge = gid >> 4;
    int c0 = gid & 15;
    if (edge < e) {
        int s = src[edge], d = dst[edge];
        float w = dinv[s] * dinv[d];
        const float* __restrict__ row = hW + (size_t)s * M;
        float* __restrict__ orow = out + (size_t)d * M;
#pragma unroll
        for (int c = c0; c < M; c += 16)
            atomic_add_f32(&orow[c], row[c] * w);
    }
}

// ---------------- launcher ----------------

extern "C" void kernel_launch(void* const* d_in, const int* in_sizes, int n_in,
                              void* d_out, int out_size, void* d_ws, size_t ws_size,
                              hipStream_t stream) {
    const float* x  = (const float*)d_in[0];
    const int*   ei = (const int*)  d_in[1];
    const float* W1 = (const float*)d_in[2];
    const float* b1 = (const float*)d_in[3];
    const float* W2 = (const float*)d_in[4];
    const float* b2 = (const float*)d_in[5];
    const float* W3 = (const float*)d_in[6];
    const float* b3 = (const float*)d_in[7];

    const int n = in_sizes[0] / DIN;     // 100000
    const int e = in_sizes[1] / 2;       // 1600000
    const int* src = ei;                 // edge_index[0]
    const int* dst = ei + e;             // edge_index[1]
    float* out = (float*)d_out;

    // workspace: dinv (n) | hW (n*DH) | h (n*DH)
    float* dinv = (float*)d_ws;
    float* hW   = dinv + n;
    float* h    = hW + (size_t)n * DH;

    const int B = 256;
    // --- normalization: deg -> dinv (in place) ---
    k_deg_init<<<(n + B - 1) / B, B, 0, stream>>>(dinv, n);
    k_deg_count<<<(e + B - 1) / B, B, 0, stream>>>(dst, dinv, e);
    k_rsqrt_inplace<<<(n + B - 1) / B, B, 0, stream>>>(dinv, n);

    const int gemmGrid = (n + 63) / 64;
    const int nm48 = n * DH, nm16 = n * DOUT;
    const int et = e * 16;

    // --- layer 1: x(32) -> 48, ReLU deferred to layer-2 load ---
    k_gemm_wmma<DIN, DH, false><<<gemmGrid, 128, 0, stream>>>(x, W1, hW, n);
    k_self_bias<DH><<<(nm48 + B - 1) / B, B, 0, stream>>>(hW, dinv, b1, h, n);
    k_scatter<DH><<<(et + B - 1) / B, B, 0, stream>>>(src, dst, hW, dinv, h, e);

    // --- layer 2: 48 -> 48 (ReLU on load), ReLU deferred to layer-3 load ---
    k_gemm_wmma<DH, DH, true><<<gemmGrid, 128, 0, stream>>>(h, W2, hW, n);
    k_self_bias<DH><<<(nm48 + B - 1) / B, B, 0, stream>>>(hW, dinv, b2, h, n);
    k_scatter<DH><<<(et + B - 1) / B, B, 0, stream>>>(src, dst, hW, dinv, h, e);

    // --- layer 3: 48 -> 16 (ReLU on load), no output ReLU, straight to d_out ---
    k_gemm_wmma<DH, DOUT, true><<<gemmGrid, 128, 0, stream>>>(h, W3, hW, n);
    k_self_bias<DOUT><<<(nm16 + B - 1) / B, B, 0, stream>>>(hW, dinv, b3, out, n);
    k_scatter<DOUT><<<(et + B - 1) / B, B, 0, stream>>>(src, dst, hW, dinv, out, e);
}